// COTREC_82102594830932
// MI455X (gfx1250) — compile-verified
//
#include <hip/hip_runtime.h>
#include <math.h>

typedef __attribute__((ext_vector_type(2))) float v2f;
typedef __attribute__((ext_vector_type(8))) float v8f;

#define EMB    112
#define NNODE  100000
#define NNZT   1000000
#define BATCH  512
#define SEQL   50

// ---------------------------------------------------------------------------
// SpMM: one wave32 per nonzero edge. Lanes 0..27 each move one float4 of the
// 112-float row (global_load_b128 gather), scale, and atomic-add scatter.
// Embedding table + accumulator are L2-resident (44.8 MB each vs 192 MB L2),
// so this is L2-atomic-rate bound, not HBM bound.
// ---------------------------------------------------------------------------
__global__ void spmm_atomic_kernel(const float* __restrict__ vals,
                                   const int*  __restrict__ rows,
                                   const int*  __restrict__ cols,
                                   const float* __restrict__ x,
                                   float* __restrict__ y, int nnz) {
    int gwave = (int)((blockIdx.x * blockDim.x + threadIdx.x) >> 5);
    int lane  = threadIdx.x & 31;
    if (gwave >= nnz) return;
    int   r = rows[gwave];
    int   c = cols[gwave];
    float v = vals[gwave];
    if (lane < EMB / 4) {
        const float4* xr = (const float4*)(x + (size_t)c * EMB) + lane;
        float*        yr = y + (size_t)r * EMB + lane * 4;
        float4 xv = *xr;
        atomicAdd(yr + 0, v * xv.x);
        atomicAdd(yr + 1, v * xv.y);
        atomicAdd(yr + 2, v * xv.z);
        atomicAdd(yr + 3, v * xv.w);
    }
}

// item_emb = (embedding + cur1 + cur2) / 3, written over cur1
__global__ void combine_kernel(const float* __restrict__ emb,
                               float* __restrict__ cur1,
                               const float* __restrict__ cur2, int n) {
    int i = blockIdx.x * blockDim.x + threadIdx.x;
    if (i < n) cur1[i] = (emb[i] + cur1[i] + cur2[i]) * (1.0f / 3.0f);
}

// sess[b,:] = mean over session items (index 0 = zero pad row); also seeds acc.
__global__ void sess_pool_kernel(const float* __restrict__ item_emb,
                                 const int*   __restrict__ session_item,
                                 const float* __restrict__ session_len,
                                 float* __restrict__ sess,
                                 float* __restrict__ accS) {
    int b = blockIdx.x;
    int k = threadIdx.x;
    if (k >= EMB) return;
    float s = 0.0f;
    for (int l = 0; l < SEQL; ++l) {
        int idx = session_item[b * SEQL + l];
        if (idx > 0) s += item_emb[(size_t)(idx - 1) * EMB + k];
    }
    s = s / session_len[b];
    sess[b * EMB + k] = s;
    accS[b * EMB + k] = s;
}

// ---------------------------------------------------------------------------
// F32 WMMA GEMM: C[M,N] = A[M,K] * B[K,N]  (TRANSB: B given as [N,K] row-major)
// TRANSB is a template parameter -> straight-line inner loop, no per-iteration
// uniform branching. K loop unrolled 4x (16 K per trip) so the compiler can
// clause the fragment loads and issue 4 V_WMMA_F32_16X16X4_F32 back-to-back.
// Block = 128 threads = 4 waves, each wave owns one 16x16 C tile along M
// (block covers 64x16 of C); all lanes active -> EXEC all 1s for WMMA.
//
// Fragment layouts (CDNA5 ISA):
//   A 16x4 f32: lane<16 -> M=lane,    K=k0+{0,1}; lane>=16 -> M=lane-16, K=k0+{2,3}
//   B 4x16 f32: symmetric, N = lane&15
//   C/D:        VGPR v -> row m0 + v + 8*(lane>>4), col n0 + (lane&15)
// ---------------------------------------------------------------------------
template <int TRANSB>
__global__ void wmma_gemm_f32_kernel(const float* __restrict__ A,
                                     const float* __restrict__ B,
                                     float* __restrict__ C,
                                     int K, int lda, int ldb, int ldc) {
    int lane = threadIdx.x & 31;
    int wv   = threadIdx.x >> 5;                 // 0..3: M sub-tile within block
    int n0   = blockIdx.x * 16;
    int m0   = (blockIdx.y * 4 + wv) * 16;
    int r    = lane & 15;
    int half = lane >> 4;

    const float* Arow = A + (size_t)(m0 + r) * lda;
    const float* Bp   = TRANSB ? (B + (size_t)(n0 + r) * ldb)   // B is [N,K]
                               : (B + (n0 + r));                // B is [K,N]

    v8f c = {};
    for (int k0 = 0; k0 < K; k0 += 16) {
#pragma unroll
        for (int s = 0; s < 4; ++s) {
            int kk = k0 + s * 4 + 2 * half;
            v2f a, b;
            a.x = Arow[kk];
            a.y = Arow[kk + 1];
            if (TRANSB) {
                b.x = Bp[kk];                    // contiguous -> global_load_b64
                b.y = Bp[kk + 1];
            } else {
                b.x = Bp[(size_t)kk * ldb];
                b.y = Bp[(size_t)(kk + 1) * ldb];
            }
            c = __builtin_amdgcn_wmma_f32_16x16x4_f32(false, a, false, b,
                                                      (short)0, c, false, false);
        }
    }

    float* Crow = C + (size_t)(m0 + 8 * half) * ldc + (n0 + r);
#pragma unroll
    for (int v = 0; v < 8; ++v)
        Crow[(size_t)v * ldc] = c[v];
}

// Row-wise L2 normalize u -> sess (next layer input) and accS += normalized.
__global__ void norm_acc_kernel(const float* __restrict__ u,
                                float* __restrict__ sess,
                                float* __restrict__ accS) {
    __shared__ float red[128];
    int b = blockIdx.x;
    int k = threadIdx.x;   // 128 threads, 112 carry data
    float v = (k < EMB) ? u[b * EMB + k] : 0.0f;
    red[k] = v * v;
    __syncthreads();
    for (int s = 64; s > 0; s >>= 1) {
        if (k < s) red[k] += red[k + s];
        __syncthreads();
    }
    float norm = fmaxf(sqrtf(red[0]), 1e-12f);
    if (k < EMB) {
        float nv = v / norm;
        sess[b * EMB + k] = nv;
        accS[b * EMB + k] += nv;
    }
}

__global__ void scale_kernel(const float* __restrict__ accS,
                             float* __restrict__ out, int n) {
    int i = blockIdx.x * blockDim.x + threadIdx.x;
    if (i < n) out[i] = accS[i] * (1.0f / 3.0f);
}

// ---------------------------------------------------------------------------
extern "C" void kernel_launch(void* const* d_in, const int* in_sizes, int n_in,
                              void* d_out, int out_size, void* d_ws, size_t ws_size,
                              hipStream_t stream) {
    const float* embedding    = (const float*)d_in[0];
    const float* adj_vals     = (const float*)d_in[1];
    const int*   adj_rows     = (const int*)d_in[2];
    const int*   adj_cols     = (const int*)d_in[3];
    const float* Dm           = (const float*)d_in[4];
    const float* Am           = (const float*)d_in[5];
    const int*   session_item = (const int*)d_in[6];
    const float* session_len  = (const float*)d_in[7];
    const float* w_sess       = (const float*)d_in[8];
    float* out = (float*)d_out;

    // Workspace layout (floats): 2 node-sized buffers + small dense scratch.
    float* cur1 = (float*)d_ws;                       // [NNODE*EMB]
    float* cur2 = cur1 + (size_t)NNODE * EMB;         // [NNODE*EMB]
    float* DA   = cur2 + (size_t)NNODE * EMB;         // [BATCH*BATCH]
    float* sess = DA   + (size_t)BATCH * BATCH;       // [BATCH*EMB]
    float* accS = sess + (size_t)BATCH * EMB;         // [BATCH*EMB]
    float* tb   = accS + (size_t)BATCH * EMB;         // [BATCH*EMB]
    float* ub   = tb   + (size_t)BATCH * EMB;         // [BATCH*EMB]

    const size_t nodeBytes = (size_t)NNODE * EMB * sizeof(float);
    const int spmmBlocks = (NNZT * 32 + 255) / 256;

    // ---- hyperconv: 2 SpMM layers ----
    hipMemsetAsync(cur1, 0, nodeBytes, stream);
    spmm_atomic_kernel<<<spmmBlocks, 256, 0, stream>>>(adj_vals, adj_rows, adj_cols,
                                                       embedding, cur1, NNZT);
    hipMemsetAsync(cur2, 0, nodeBytes, stream);
    spmm_atomic_kernel<<<spmmBlocks, 256, 0, stream>>>(adj_vals, adj_rows, adj_cols,
                                                       cur1, cur2, NNZT);
    int ncomb = NNODE * EMB;
    combine_kernel<<<(ncomb + 255) / 256, 256, 0, stream>>>(embedding, cur1, cur2, ncomb);

    // ---- session mean pooling (seeds acc) ----
    sess_pool_kernel<<<BATCH, 128, 0, stream>>>(cur1, session_item, session_len,
                                                sess, accS);

    // ---- DA = D @ A  (512x512x512, f32 WMMA) ----
    wmma_gemm_f32_kernel<0><<<dim3(BATCH / 16, BATCH / 64), 128, 0, stream>>>(
        Dm, Am, DA, BATCH, BATCH, BATCH, BATCH);

    // ---- 2 session-conv layers ----
    for (int layer = 0; layer < 2; ++layer) {
        const float* w = w_sess + (size_t)layer * EMB * EMB;
        // t = sess @ w^T   (M=512, N=112, K=112; B supplied as [N,K])
        wmma_gemm_f32_kernel<1><<<dim3(EMB / 16, BATCH / 64), 128, 0, stream>>>(
            sess, w, tb, EMB, EMB, EMB, EMB);
        // u = DA @ t       (M=512, N=112, K=512)
        wmma_gemm_f32_kernel<0><<<dim3(EMB / 16, BATCH / 64), 128, 0, stream>>>(
            DA, tb, ub, BATCH, BATCH, EMB, EMB);
        // sess = l2norm(u); acc += sess
        norm_acc_kernel<<<BATCH, 128, 0, stream>>>(ub, sess, accS);
    }

    // ---- out = acc / 3 ----
    scale_kernel<<<(BATCH * EMB + 255) / 256, 256, 0, stream>>>(accS, out, BATCH * EMB);
}